// GCN_21509196218552
// MI455X (gfx1250) — compile-verified
//
#include <hip/hip_runtime.h>
#include <hip/hip_bf16.h>

typedef float v2f __attribute__((ext_vector_type(2)));
typedef float v8f __attribute__((ext_vector_type(8)));

#define HID 64
#define NGRAPH 128
#define DENSE 128

// ---------------------------------------------------------------------------
// Hardware f32 atomic add: force global_atomic_add_f32 (no CAS loop).
// ---------------------------------------------------------------------------
__device__ __forceinline__ void atomAddF(float* p, float v) {
    asm volatile("global_atomic_add_f32 %0, %1, off scope:SCOPE_DEV"
                 :: "v"(p), "v"(v) : "memory");
}

// ---------------------------------------------------------------------------
// Degree / normalization
// ---------------------------------------------------------------------------
__global__ void init_deg_kernel(int* __restrict__ degi, int N) {
    int i = blockIdx.x * blockDim.x + threadIdx.x;
    if (i < N) degi[i] = 1;                 // self-loop contributes 1
}

__global__ void deg_kernel(const int* __restrict__ dst, int* __restrict__ degi, int E) {
    int e = blockIdx.x * blockDim.x + threadIdx.x;
    if (e < E) atomicAdd(&degi[dst[e]], 1);
}

__global__ void dinv_kernel(const int* __restrict__ degi, float* __restrict__ dinv, int N) {
    int i = blockIdx.x * blockDim.x + threadIdx.x;
    if (i < N) {
        float d = (float)degi[i];
        dinv[i] = d > 0.0f ? rsqrtf(d) : 0.0f;
    }
}

// ---------------------------------------------------------------------------
// T[m, 0:64] = dinv[m] * (Hin[m, 0:K] @ W[0:K, 0:64])   via V_WMMA_F32_16X16X4_F32
// K is a compile-time constant -> fully unrolled, bfrag in fixed VGPRs,
// no divergent loads, EXEC all-ones at every WMMA.
// Block = 128 threads = 4 waves; wave w owns output cols [16w, 16w+16).
// ---------------------------------------------------------------------------
template <int K>
__global__ void wmma_gemm64_kernel(const float* __restrict__ Hin,
                                   const float* __restrict__ W,
                                   const float* __restrict__ dinv,
                                   float* __restrict__ Tout, int N) {
    constexpr int KT = (K + 3) / 4;           // K=3 -> 1, K=64 -> 16
    const int lane = threadIdx.x & 31;
    const int wave = threadIdx.x >> 5;        // 0..3
    const int col0 = wave * 16;
    const int l    = lane & 15;
    const int hoff = (lane >> 4) << 1;        // 0 (lanes 0-15) or 2 (lanes 16-31)
    const bool hi  = (lane & 16) != 0;

    // Preload B fragments: lanes 0-15 rows k,k+1 ; lanes 16-31 rows k+2,k+3
    v2f bfrag[KT];
    #pragma unroll
    for (int i = 0; i < KT; ++i) {
        const int k = i * 4 + hoff;
        v2f b;
        if constexpr (K % 4 == 0) {
            b.x = W[(size_t)k * HID + col0 + l];
            b.y = W[(size_t)(k + 1) * HID + col0 + l];
        } else {
            b.x = (k     < K) ? W[(size_t)k       * HID + col0 + l] : 0.0f;
            b.y = (k + 1 < K) ? W[(size_t)(k + 1) * HID + col0 + l] : 0.0f;
        }
        bfrag[i] = b;
    }

    const int ntiles = (N + 15) >> 4;
    for (int tile = blockIdx.x; tile < ntiles; tile += gridDim.x) {
        const int row0 = tile << 4;
        // Clamp (instead of predicating) A-row loads for a possible partial tile.
        const int arow = min(row0 + l, N - 1);
        const float* hrow = Hin + (size_t)arow * K;

        v8f acc = {0.f, 0.f, 0.f, 0.f, 0.f, 0.f, 0.f, 0.f};
        #pragma unroll
        for (int i = 0; i < KT; ++i) {
            v2f a;
            if constexpr (K % 4 == 0) {
                a = *(const v2f*)(hrow + i * 4 + hoff);   // 8B-aligned vector load
            } else {
                const int k = i * 4 + hoff;
                a.x = (k     < K) ? hrow[k]     : 0.0f;
                a.y = (k + 1 < K) ? hrow[k + 1] : 0.0f;
            }
            acc = __builtin_amdgcn_wmma_f32_16x16x4_f32(
                false, a, false, bfrag[i], (short)0, acc, false, false);
        }

        // D layout: VGPR v -> row row0+v (lanes 0-15) / row0+v+8 (lanes 16-31);
        // lane%16 -> column. Scale by dinv[row] on the way out.
        const int mbase = row0 + (hi ? 8 : 0);
        if (row0 + 16 <= N) {                 // fast path (always, when N%16==0)
            #pragma unroll
            for (int v = 0; v < 8; ++v) {
                const int m = mbase + v;
                Tout[(size_t)m * HID + col0 + l] = acc[v] * dinv[m];
            }
        } else {
            #pragma unroll
            for (int v = 0; v < 8; ++v) {
                const int m = mbase + v;
                if (m < N) Tout[(size_t)m * HID + col0 + l] = acc[v] * dinv[m];
            }
        }
    }
}

// ---------------------------------------------------------------------------
// Edge scatter: A[dst] += T[src]   (16 threads per edge, float4 per thread)
// ---------------------------------------------------------------------------
__global__ void edge_scatter_kernel(const int* __restrict__ src, const int* __restrict__ dst,
                                    const float* __restrict__ T, float* __restrict__ A, int E) {
    long long t = (long long)blockIdx.x * blockDim.x + threadIdx.x;
    int e = (int)(t >> 4);
    if (e >= E) return;
    int c = (int)(t & 15);
    int s = src[e];
    int d = dst[e];
    float4 v = ((const float4*)T)[(size_t)s * 16 + c];
    float* out = A + (size_t)d * HID + c * 4;
    atomAddF(out + 0, v.x);
    atomAddF(out + 1, v.y);
    atomAddF(out + 2, v.z);
    atomAddF(out + 3, v.w);
}

// ---------------------------------------------------------------------------
// H[i] = relu(dinv[i] * (A[i] + T[i]) + b)   (T[i] term = folded self-loop)
// ---------------------------------------------------------------------------
__global__ void finalize_kernel(const float* __restrict__ A, const float* __restrict__ T,
                                const float* __restrict__ dinv, const float* __restrict__ b,
                                float* __restrict__ H, int N) {
    long long t = (long long)blockIdx.x * blockDim.x + threadIdx.x;
    int i = (int)(t >> 4);
    if (i >= N) return;
    int c = (int)(t & 15) * 4;
    float di = dinv[i];
    size_t base = (size_t)i * HID + c;
    float4 a  = *(const float4*)(A + base);
    float4 tt = *(const float4*)(T + base);
    float4 r;
    r.x = fmaxf(di * (a.x + tt.x) + b[c + 0], 0.0f);
    r.y = fmaxf(di * (a.y + tt.y) + b[c + 1], 0.0f);
    r.z = fmaxf(di * (a.z + tt.z) + b[c + 2], 0.0f);
    r.w = fmaxf(di * (a.w + tt.w) + b[c + 3], 0.0f);
    *(float4*)(H + base) = r;
}

// ---------------------------------------------------------------------------
// Last layer: finalize fused with global mean-pool accumulation (H never stored)
// ---------------------------------------------------------------------------
__global__ void finalize_pool_kernel(const float* __restrict__ A, const float* __restrict__ T,
                                     const float* __restrict__ dinv, const float* __restrict__ b,
                                     const int* __restrict__ batch,
                                     float* __restrict__ sums, int* __restrict__ cnts, int N) {
    long long t = (long long)blockIdx.x * blockDim.x + threadIdx.x;
    int i = (int)(t >> 4);
    if (i >= N) return;
    int c = (int)(t & 15) * 4;
    float di = dinv[i];
    size_t base = (size_t)i * HID + c;
    float4 a  = *(const float4*)(A + base);
    float4 tt = *(const float4*)(T + base);
    int g = batch[i];
    float* s = sums + (size_t)g * HID + c;
    atomAddF(s + 0, fmaxf(di * (a.x + tt.x) + b[c + 0], 0.0f));
    atomAddF(s + 1, fmaxf(di * (a.y + tt.y) + b[c + 1], 0.0f));
    atomAddF(s + 2, fmaxf(di * (a.z + tt.z) + b[c + 2], 0.0f));
    atomAddF(s + 3, fmaxf(di * (a.w + tt.w) + b[c + 3], 0.0f));
    if (c == 0) atomicAdd(&cnts[g], 1);
}

// ---------------------------------------------------------------------------
// MLP head: out[g] = relu(pooled @ fc1 + b1) @ fc2 + b2   (one thread per graph)
// ---------------------------------------------------------------------------
__global__ void head_kernel(const float* __restrict__ sums, const int* __restrict__ cnts,
                            const float* __restrict__ fc1w, const float* __restrict__ fc1b,
                            const float* __restrict__ fc2w, const float* __restrict__ fc2b,
                            float* __restrict__ out) {
    int g = threadIdx.x;
    if (g >= NGRAPH) return;
    float pooled[HID];
    float inv = 1.0f / fmaxf((float)cnts[g], 1.0f);
    #pragma unroll
    for (int c = 0; c < HID; ++c) pooled[c] = sums[(size_t)g * HID + c] * inv;
    float o = 0.0f;
    for (int j = 0; j < DENSE; ++j) {
        float acc = fc1b[j];
        #pragma unroll
        for (int c = 0; c < HID; ++c) acc = fmaf(pooled[c], fc1w[(size_t)c * DENSE + j], acc);
        o = fmaf(fmaxf(acc, 0.0f), fc2w[j], o);
    }
    out[g] = o + fc2b[0];
}

// ---------------------------------------------------------------------------
extern "C" void kernel_launch(void* const* d_in, const int* in_sizes, int n_in,
                              void* d_out, int out_size, void* d_ws, size_t ws_size,
                              hipStream_t stream) {
    const float* x     = (const float*)d_in[0];
    const int*   ei    = (const int*)  d_in[1];
    const int*   batch = (const int*)  d_in[2];
    const float* W0    = (const float*)d_in[3];
    const float* b0    = (const float*)d_in[4];
    const float* W1    = (const float*)d_in[5];
    const float* b1    = (const float*)d_in[6];
    const float* W2    = (const float*)d_in[7];
    const float* b2    = (const float*)d_in[8];
    const float* fc1w  = (const float*)d_in[9];
    const float* fc1b  = (const float*)d_in[10];
    const float* fc2w  = (const float*)d_in[11];
    const float* fc2b  = (const float*)d_in[12];

    const int N = in_sizes[0] / 3;
    const int E = in_sizes[1] / 2;
    const int* src = ei;
    const int* dst = ei + E;

    // Workspace carve-up
    const size_t N64 = (size_t)N * HID;
    float* T    = (float*)d_ws;
    float* A    = T + N64;
    float* H    = A + N64;
    float* dinv = H + N64;
    int*   degi = (int*)(dinv + N);
    float* sums = (float*)(degi + N);
    int*   cnts = (int*)(sums + (size_t)NGRAPH * HID);

    const int TPB = 256;
    // Degree + D^-1/2
    init_deg_kernel<<<(N + TPB - 1) / TPB, TPB, 0, stream>>>(degi, N);
    deg_kernel<<<(E + TPB - 1) / TPB, TPB, 0, stream>>>(dst, degi, E);
    dinv_kernel<<<(N + TPB - 1) / TPB, TPB, 0, stream>>>(degi, dinv, N);

    const int edgeBlocks = (int)(((long long)E * 16 + TPB - 1) / TPB);
    const int nodeBlocks = (int)(((long long)N * 16 + TPB - 1) / TPB);

    // --- layer 0: K=3 ---
    wmma_gemm64_kernel<3><<<2048, 128, 0, stream>>>(x, W0, dinv, T, N);
    hipMemsetAsync(A, 0, N64 * sizeof(float), stream);
    edge_scatter_kernel<<<edgeBlocks, TPB, 0, stream>>>(src, dst, T, A, E);
    finalize_kernel<<<nodeBlocks, TPB, 0, stream>>>(A, T, dinv, b0, H, N);

    // --- layer 1: K=64 ---
    wmma_gemm64_kernel<64><<<2048, 128, 0, stream>>>(H, W1, dinv, T, N);
    hipMemsetAsync(A, 0, N64 * sizeof(float), stream);
    edge_scatter_kernel<<<edgeBlocks, TPB, 0, stream>>>(src, dst, T, A, E);
    finalize_kernel<<<nodeBlocks, TPB, 0, stream>>>(A, T, dinv, b1, H, N);

    // --- layer 2: K=64, finalize fused with mean-pool ---
    wmma_gemm64_kernel<64><<<2048, 128, 0, stream>>>(H, W2, dinv, T, N);
    hipMemsetAsync(A, 0, N64 * sizeof(float), stream);
    hipMemsetAsync(sums, 0, (size_t)NGRAPH * HID * sizeof(float), stream);
    hipMemsetAsync(cnts, 0, (size_t)NGRAPH * sizeof(int), stream);
    edge_scatter_kernel<<<edgeBlocks, TPB, 0, stream>>>(src, dst, T, A, E);
    finalize_pool_kernel<<<nodeBlocks, TPB, 0, stream>>>(A, T, dinv, b2, batch, sums, cnts, N);

    // --- head ---
    head_kernel<<<1, 128, 0, stream>>>(sums, cnts, fc1w, fc1b, fc2w, fc2b, (float*)d_out);
}